// LocalSpatialSimilarity_34875134443852
// MI455X (gfx1250) — compile-verified
//
#include <hip/hip_runtime.h>
#include <hip/hip_bf16.h>
#include <math.h>

// Problem constants (from reference): T=4, B=32, C=256, H=W=32.
#define T_   4
#define B_   32
#define C_   256
#define HW_  1024            // H*W pixels per batch
#define CSPLIT 8             // channel slices (32 channels each) for parallelism
#define CG_PER_SLICE 8       // 32 channels / K(=4) per WMMA

typedef float v2f __attribute__((ext_vector_type(2)));
typedef float v8f __attribute__((ext_vector_type(8)));

// ---------------------------------------------------------------------------
// Kernel 1: streaming reduction over (T, channel-slice) using WMMA f32 16x16x4.
//   Per wave: 16 pixels of one batch b, channels [slice*32, slice*32+32).
//   Lane layout matches the f32 WMMA A/B striping:
//     lane l<16  : pixel m=l,   holds channels 4g+0 (vgpr0) and 4g+1 (vgpr1)
//     lane l>=16 : pixel m=l-16,holds channels 4g+2 (vgpr0) and 4g+3 (vgpr1)
//   gram = Xm^T * Xm  -> diagonal = sum_c xm^2 per pixel
//   cs   = ones * Xm  -> every row = sum_c xm  per pixel (read from acc VGPR0)
// ---------------------------------------------------------------------------
__global__ __launch_bounds__(256)
void lss_partial_kernel(const float* __restrict__ x,
                        float* __restrict__ pcs,   // [CSPLIT][B][HW]
                        float* __restrict__ psq) { // [CSPLIT][B][HW]
  const int lane  = threadIdx.x & 31;
  const int wid   = threadIdx.x >> 5;
  const int pg    = blockIdx.x * 8 + wid;   // pixel group 0..63
  const int b     = blockIdx.y;
  const int slice = blockIdx.z;
  const int m     = lane & 15;              // pixel within group
  const int half  = lane >> 4;              // 0: K=0/1, 1: K=2/3
  const int p     = pg * 16 + m;

  const size_t tstride = (size_t)B_ * C_ * HW_;

  v8f gram = {0.f,0.f,0.f,0.f,0.f,0.f,0.f,0.f};
  v8f cs   = {0.f,0.f,0.f,0.f,0.f,0.f,0.f,0.f};
  const v2f ones = {1.f, 1.f};

  #pragma unroll
  for (int g = 0; g < CG_PER_SLICE; ++g) {
    const int c0 = slice * 32 + g * 4 + half * 2;
    const float* ptr = x + ((size_t)b * C_ + c0) * HW_ + p;
    float s0 = 0.f, s1 = 0.f;
    #pragma unroll
    for (int t = 0; t < T_; ++t) {
      s0 += ptr[(size_t)t * tstride];          // channel c0
      s1 += ptr[(size_t)t * tstride + HW_];    // channel c0+1
    }
    v2f xm = { 0.25f * s0, 0.25f * s1 };       // mean over T
    // D = A*B + C ; A == B == xm tile -> Gram (diag = sum of squares)
    gram = __builtin_amdgcn_wmma_f32_16x16x4_f32(
        false, xm, false, xm, (short)0, gram, false, false);
    // A = ones -> every row of D is the per-pixel channel sum
    cs   = __builtin_amdgcn_wmma_f32_16x16x4_f32(
        false, ones, false, xm, (short)0, cs, false, false);
  }

  // Diagonal element (m,m) lives at lane (m<8 ? m : m+16), acc register (m&7).
  const int src = (m < 8) ? m : (m + 16);
  float sumsq = 0.f;
  #pragma unroll
  for (int r = 0; r < 8; ++r) {
    float tv = __shfl(gram[r], src, 32);
    if ((m & 7) == r) sumsq = tv;
  }

  if (lane < 16) {
    const size_t o = ((size_t)slice * B_ + b) * HW_ + p;
    pcs[o] = cs[0];     // acc VGPR0, lane n  == csum of pixel n (row M=0)
    psq[o] = sumsq;
  }
}

// ---------------------------------------------------------------------------
// Kernel 2: per-batch finalize. Sum the CSPLIT partials (fixed order ->
// deterministic), 3x3 box filter via LDS, cosine-sim closed form, masked
// softmax over the 1024 pixels of the block.
// ---------------------------------------------------------------------------
__global__ __launch_bounds__(1024)
void lss_finalize_kernel(const float* __restrict__ pcs,
                         const float* __restrict__ psq,
                         const unsigned char* __restrict__ mask, // bool [B][HW]
                         float* __restrict__ out) {              // [B][H][W]
  __shared__ float s_csum[HW_];
  __shared__ float s_red[32];

  const int p = threadIdx.x;     // pixel 0..1023
  const int b = blockIdx.x;

  float csum = 0.f, sumsq = 0.f;
  #pragma unroll
  for (int s = 0; s < CSPLIT; ++s) {
    const size_t o = ((size_t)s * B_ + b) * HW_ + p;
    csum  += pcs[o];
    sumsq += psq[o];
  }
  s_csum[p] = csum;
  __syncthreads();

  // 3x3 zero-padded box sum of csum, /9
  const int px = p & 31, py = p >> 5;
  float lmsum = 0.f;
  #pragma unroll
  for (int dy = -1; dy <= 1; ++dy) {
    const int yy = py + dy;
    if (yy < 0 || yy > 31) continue;
    #pragma unroll
    for (int dx = -1; dx <= 1; ++dx) {
      const int xx = px + dx;
      if (xx < 0 || xx > 31) continue;
      lmsum += s_csum[yy * 32 + xx];
    }
  }
  const float lm   = lmsum * (1.f / 9.f);
  const float dotv = lm * csum;
  const float nx   = sqrtf(sumsq);
  const float ny   = 16.f * fabsf(lm);         // sqrt(C)=16
  const float EPS  = 1e-6f;
  const float sim  = dotv / (fmaxf(nx, EPS) * fmaxf(ny, EPS));

  const bool masked = mask[(size_t)b * HW_ + p] != 0;
  const float score = masked ? -__builtin_inff() : -sim;

  const int lane = p & 31, wid = p >> 5;

  // --- block max ---
  float mx = score;
  #pragma unroll
  for (int off = 16; off; off >>= 1) mx = fmaxf(mx, __shfl_xor(mx, off, 32));
  if (lane == 0) s_red[wid] = mx;
  __syncthreads();
  if (wid == 0) {
    float v = s_red[lane];
    #pragma unroll
    for (int off = 16; off; off >>= 1) v = fmaxf(v, __shfl_xor(v, off, 32));
    if (lane == 0) s_red[0] = v;
  }
  __syncthreads();
  const float gmax = s_red[0];
  __syncthreads();

  // --- exp + block sum ---
  const float e = masked ? 0.f : __expf(score - gmax);
  float sm = e;
  #pragma unroll
  for (int off = 16; off; off >>= 1) sm += __shfl_xor(sm, off, 32);
  if (lane == 0) s_red[wid] = sm;
  __syncthreads();
  if (wid == 0) {
    float v = s_red[lane];
    #pragma unroll
    for (int off = 16; off; off >>= 1) v += __shfl_xor(v, off, 32);
    if (lane == 0) s_red[0] = v;
  }
  __syncthreads();
  const float gsum = s_red[0];

  out[(size_t)b * HW_ + p] = e / gsum;
}

// ---------------------------------------------------------------------------
extern "C" void kernel_launch(void* const* d_in, const int* in_sizes, int n_in,
                              void* d_out, int out_size, void* d_ws, size_t ws_size,
                              hipStream_t stream) {
  const float*         x    = (const float*)d_in[0];          // [T,B,C,H,W] f32
  const unsigned char* mask = (const unsigned char*)d_in[1];  // bool [B,HW]
  float*               out  = (float*)d_out;                  // [B,H,W] f32

  // Workspace: two [CSPLIT][B][HW] f32 arrays = 2 MiB total.
  float* pcs = (float*)d_ws;
  float* psq = pcs + (size_t)CSPLIT * B_ * HW_;

  dim3 g1(8 /*pixel-group blocks*/, B_, CSPLIT);
  lss_partial_kernel<<<g1, 256, 0, stream>>>(x, pcs, psq);
  lss_finalize_kernel<<<B_, 1024, 0, stream>>>(pcs, psq, mask, out);
}